// HardVoxelizer_8100308320785
// MI455X (gfx1250) — compile-verified
//
#include <hip/hip_runtime.h>
#include <stdint.h>

// ---------------------------------------------------------------------------
// HardVoxelizer for MI455X (gfx1250).
// Bandwidth/atomic-bound binning problem: points (25MB) -> 120000x32x3 voxels.
// Voxel grid state (3.93M cells, ~16MB) is L2-resident (192MB L2).
// CDNA5-specific paths used: GLOBAL_LOAD_ASYNC_TO_LDS_B32 + s_wait_asynccnt
// for point staging, LDS atomics / LDS scans for the occupancy prefix sum.
// ---------------------------------------------------------------------------

#define GXD 512
#define GYD 512
#define GZD 15
#define NVOX (GXD * GYD * GZD)          /* 3,932,160 */
#define NPTS (8 * 262144)               /* 2,097,152 */
#define MAXVOX 120000
#define MAXPPV 32
#define VOX_ELEMS (MAXVOX * MAXPPV * 3) /* 11,520,000 */
#define COORS_BASE VOX_ELEMS
#define NP_BASE (COORS_BASE + MAXVOX * 3)
#define OUT_TOTAL (NP_BASE + MAXVOX)    /* 12,000,000 */

#define SCAN_CHUNK 4096
#define SCAN_BLOCKS (NVOX / SCAN_CHUNK) /* 960 exactly */
#define PER_THREAD (SCAN_CHUNK / 256)   /* 16 */

// ---------------------------------------------------------------------------
// Kernel 1: initialize all state we read/write this call (ws is poisoned 0xAA
// once; d_out must be rebuilt every call).
// ---------------------------------------------------------------------------
__global__ void voxel_init_kernel(float* __restrict__ out,
                                  int* __restrict__ head,
                                  int* __restrict__ vox2lin) {
  const int stride = gridDim.x * blockDim.x;
  int g = blockIdx.x * blockDim.x + threadIdx.x;
  for (int i = g; i < OUT_TOTAL; i += stride) {
    float v = 0.0f;
    if (i >= COORS_BASE && i < NP_BASE) v = -1.0f;  // coors init to -1
    out[i] = v;
  }
  for (int i = g; i < NVOX; i += stride) head[i] = -1;
  for (int i = g; i < MAXVOX; i += stride) vox2lin[i] = -1;
}

// ---------------------------------------------------------------------------
// Kernel 2: scatter. Stage 256 points (3KB) into LDS with CDNA5 async
// global->LDS loads, then bin each point and push its index onto the
// per-voxel linked list (atomicExch). Push order is nondeterministic but the
// fill kernel sorts indices, so results are deterministic.
// ---------------------------------------------------------------------------
__global__ void voxel_scatter_kernel(const float* __restrict__ points,
                                     int* __restrict__ head,
                                     int* __restrict__ next) {
  __shared__ float tile[256 * 3];
  const int tid = threadIdx.x;
  const int pidx = blockIdx.x * 256 + tid;  // NPTS % 256 == 0, no tail

  // Async copy of this lane's 12 bytes into LDS (3 x b32).
  // Low 32 bits of a generic LDS pointer are the wave-relative LDS byte addr.
  unsigned lds_addr = (unsigned)(uintptr_t)(&tile[3 * tid]);
  unsigned gbyte = (unsigned)(pidx) * 12u;  // 25MB total < 2^31, fits i32
  asm volatile(
      "global_load_async_to_lds_b32 %0, %1, %2 offset:0\n\t"
      "global_load_async_to_lds_b32 %0, %1, %2 offset:4\n\t"
      "global_load_async_to_lds_b32 %0, %1, %2 offset:8\n\t"
      :
      : "v"(lds_addr), "v"(gbyte), "s"(points)
      : "memory");
  asm volatile("s_wait_asynccnt 0" ::: "memory");
  // Each lane consumes only the bytes it staged itself -> no barrier needed.

  const float px = tile[3 * tid + 0];
  const float py = tile[3 * tid + 1];
  const float pz = tile[3 * tid + 2];

  // floor((p - lo) / vs); NaN fails every compare -> invalid, matching the
  // reference's not_nan_mask & in_range logic.
  const float fx = floorf((px + 51.2f) / 0.2f);
  const float fy = floorf((py + 51.2f) / 0.2f);
  const float fz = floorf((pz + 3.0f) / 0.4f);
  const bool valid = (fx >= 0.0f) && (fx < 512.0f) &&
                     (fy >= 0.0f) && (fy < 512.0f) &&
                     (fz >= 0.0f) && (fz < 15.0f);
  if (valid) {
    const int cx = (int)fx, cy = (int)fy, cz = (int)fz;
    const int lin = (cz * GYD + cy) * GXD + cx;
    const int old = atomicExch(&head[lin], pidx);
    next[pidx] = old;
  }
}

// ---------------------------------------------------------------------------
// Kernel 3: per-block occupied-voxel counts (for the ascending-lin ordinal).
// ---------------------------------------------------------------------------
__global__ void voxel_blocksum_kernel(const int* __restrict__ head,
                                      int* __restrict__ bsums) {
  __shared__ int ssum;
  const int tid = threadIdx.x;
  if (tid == 0) ssum = 0;
  __syncthreads();
  const int base = blockIdx.x * SCAN_CHUNK + tid * PER_THREAD;
  int c = 0;
#pragma unroll
  for (int k = 0; k < PER_THREAD; ++k) c += (head[base + k] != -1) ? 1 : 0;
  atomicAdd(&ssum, c);
  __syncthreads();
  if (tid == 0) bsums[blockIdx.x] = ssum;
}

// ---------------------------------------------------------------------------
// Kernel 4: exclusive scan of the 960 block sums (single block, LDS serial —
// trivially cheap at this size).
// ---------------------------------------------------------------------------
__global__ void voxel_scan_kernel(int* __restrict__ bsums) {
  __shared__ int s[SCAN_BLOCKS];
  const int t = threadIdx.x;
  if (t < SCAN_BLOCKS) s[t] = bsums[t];
  __syncthreads();
  if (t == 0) {
    int run = 0;
    for (int i = 0; i < SCAN_BLOCKS; ++i) {
      const int v = s[i];
      s[i] = run;
      run += v;
    }
  }
  __syncthreads();
  if (t < SCAN_BLOCKS) bsums[t] = s[t];
}

// ---------------------------------------------------------------------------
// Kernel 5: assign ascending-lin ordinals; ordinals < MAXVOX record their lin
// id into vox2lin (the kept-bucket table).
// ---------------------------------------------------------------------------
__global__ void voxel_emit_kernel(const int* __restrict__ head,
                                  const int* __restrict__ boffsets,
                                  int* __restrict__ vox2lin) {
  __shared__ int tsum[256];
  const int tid = threadIdx.x;
  const int base = blockIdx.x * SCAN_CHUNK + tid * PER_THREAD;
  int c = 0;
#pragma unroll
  for (int k = 0; k < PER_THREAD; ++k) c += (head[base + k] != -1) ? 1 : 0;
  tsum[tid] = c;
  __syncthreads();
  if (tid == 0) {
    int run = 0;
    for (int i = 0; i < 256; ++i) {
      const int v = tsum[i];
      tsum[i] = run;
      run += v;
    }
  }
  __syncthreads();
  int ord = boffsets[blockIdx.x] + tsum[tid];
  for (int k = 0; k < PER_THREAD; ++k) {
    const int lin = base + k;
    if (head[lin] != -1) {
      if (ord < MAXVOX) vox2lin[ord] = lin;
      ++ord;
    }
  }
}

// ---------------------------------------------------------------------------
// Kernel 6: fill. One thread per kept voxel: traverse the linked list keeping
// the 32 smallest point indices in ascending order (== stable-sort rank),
// then gather the point rows and write voxels / coors(z,y,x) / num_points.
// d_out is float* for the whole concatenated tuple.
// ---------------------------------------------------------------------------
__global__ void voxel_fill_kernel(const float* __restrict__ points,
                                  const int* __restrict__ head,
                                  const int* __restrict__ next,
                                  const int* __restrict__ vox2lin,
                                  float* __restrict__ out) {
  const int vid = blockIdx.x * blockDim.x + threadIdx.x;
  if (vid >= MAXVOX) return;
  const int lin = vox2lin[vid];
  if (lin < 0) return;  // fewer than MAXVOX occupied voxels; defaults stand

  int idxs[MAXPPV];
  int kept = 0;
  int count = 0;
  int cur = head[lin];
  while (cur != -1) {
    ++count;
    if (kept < MAXPPV) {
      int j = kept++;
      while (j > 0 && idxs[j - 1] > cur) {
        idxs[j] = idxs[j - 1];
        --j;
      }
      idxs[j] = cur;
    } else if (cur < idxs[MAXPPV - 1]) {
      int j = MAXPPV - 1;
      while (j > 0 && idxs[j - 1] > cur) {
        idxs[j] = idxs[j - 1];
        --j;
      }
      idxs[j] = cur;
    }
    cur = next[cur];
  }

  const int x = lin & (GXD - 1);
  const int y = (lin >> 9) & (GYD - 1);
  const int z = lin >> 18;
  out[COORS_BASE + vid * 3 + 0] = (float)z;
  out[COORS_BASE + vid * 3 + 1] = (float)y;
  out[COORS_BASE + vid * 3 + 2] = (float)x;
  out[NP_BASE + vid] = (float)(count < MAXPPV ? count : MAXPPV);

  const int vbase = vid * (MAXPPV * 3);
  for (int r = 0; r < kept; ++r) {
    const float* __restrict__ s = points + (size_t)idxs[r] * 3;
    out[vbase + r * 3 + 0] = s[0];
    out[vbase + r * 3 + 1] = s[1];
    out[vbase + r * 3 + 2] = s[2];
  }
}

// ---------------------------------------------------------------------------
extern "C" void kernel_launch(void* const* d_in, const int* in_sizes, int n_in,
                              void* d_out, int out_size, void* d_ws,
                              size_t ws_size, hipStream_t stream) {
  (void)in_sizes;
  (void)n_in;
  (void)out_size;
  (void)ws_size;
  const float* points = (const float*)d_in[0];
  float* out = (float*)d_out;

  // Workspace layout (~24.7 MB total)
  int* head = (int*)d_ws;                 // NVOX
  int* next = head + NVOX;                // NPTS
  int* bsums = next + NPTS;               // SCAN_BLOCKS, padded to 1024
  int* vox2lin = bsums + 1024;            // MAXVOX

  voxel_init_kernel<<<1024, 256, 0, stream>>>(out, head, vox2lin);
  voxel_scatter_kernel<<<NPTS / 256, 256, 0, stream>>>(points, head, next);
  voxel_blocksum_kernel<<<SCAN_BLOCKS, 256, 0, stream>>>(head, bsums);
  voxel_scan_kernel<<<1, 1024, 0, stream>>>(bsums);
  voxel_emit_kernel<<<SCAN_BLOCKS, 256, 0, stream>>>(head, bsums, vox2lin);
  voxel_fill_kernel<<<(MAXVOX + 255) / 256, 256, 0, stream>>>(points, head,
                                                              next, vox2lin,
                                                              out);
}